// HierarchicalPointEncoder_23175643529478
// MI455X (gfx1250) — compile-verified
//
#include <hip/hip_runtime.h>
#include <hip/hip_bf16.h>

// ---------------------------------------------------------------------------
// MI455X / gfx1250, wave32. All matrix math via V_WMMA_F32_16X16X4_F32.
// ---------------------------------------------------------------------------

typedef float v2f __attribute__((ext_vector_type(2)));
typedef float v8f __attribute__((ext_vector_type(8)));

__device__ __forceinline__ v8f wmma_f32_16x16x4(v2f a, v2f b, v8f c) {
  // 8 args: (neg_a, A, neg_b, B, c_mod, C, reuse_a, reuse_b)
  return __builtin_amdgcn_wmma_f32_16x16x4_f32(false, a, false, b, (short)0, c,
                                               false, false);
}

// Wave-local LDS ordering: lowers to s_wait_dscnt, no s_barrier.
__device__ __forceinline__ void lds_fence_wave() {
  __builtin_amdgcn_fence(__ATOMIC_ACQ_REL, "workgroup", "local");
}

#define TKN 16          // neighbors
#define WAVES_PER_BLK 8 // 256 threads

// ---------------------------------------------------------------------------
// Input projection: coor extraction, squared norms, f0 = coor @ W_in^T + b_in
// ---------------------------------------------------------------------------
__global__ void k_proj(const float* __restrict__ x, const float* __restrict__ Win,
                       const float* __restrict__ bin, float* __restrict__ coor,
                       float* __restrict__ f0, float* __restrict__ sqn, int total) {
  int i = blockIdx.x * blockDim.x + threadIdx.x;
  if (i >= total) return;
  const float* xi = x + (size_t)i * 7;
  float cx = xi[0], cy = xi[1], cz = xi[2];
  coor[(size_t)i * 3 + 0] = cx;
  coor[(size_t)i * 3 + 1] = cy;
  coor[(size_t)i * 3 + 2] = cz;
  sqn[i] = cx * cx + cy * cy + cz * cz;
#pragma unroll
  for (int o = 0; o < 8; ++o) {
    f0[(size_t)i * 8 + o] =
        cx * Win[o * 3 + 0] + cy * Win[o * 3 + 1] + cz * Win[o * 3 + 2] + bin[o];
  }
}

// ---------------------------------------------------------------------------
// kNN: each wave owns 16 queries; loops key tiles of 16.
// dot tile via WMMA f32 16x16x4 (coords padded to K=4 with 0).
// selection metric: |k|^2 - 2*q.k  (|q|^2 is constant per row -> drop).
// Lanes 0-15 keep a register top-16 (value+index) with replace-max.
// Branchless fragment loads; wave-private LDS tile (fence, no block barrier).
// ---------------------------------------------------------------------------
__global__ void k_knn(const float* __restrict__ cq, const float* __restrict__ ck,
                      const float* __restrict__ kn, int* __restrict__ knn_idx,
                      int Q, int N) {
  __shared__ float tile[WAVES_PER_BLK][16][16];
  const int lane = threadIdx.x & 31;
  const int wave = threadIdx.x >> 5;
  const int gw = blockIdx.x * WAVES_PER_BLK + wave;
  const int qtPerBatch = Q >> 4;
  const int b = gw / qtPerBatch;
  const int qt = gw % qtPerBatch;
  const int m = lane & 15;
  const int hf = lane >> 4;
  const int q = qt * 16 + m;

  const float* cqb = cq + (size_t)b * Q * 3;
  const float* ckb = ck + (size_t)b * N * 3;
  const float* knb = kn + (size_t)b * N;

  // A fragment: rows = queries. VGPR0 = {K0 | K2}, VGPR1 = {K1 | K3(pad 0)}
  // hf==0 reads (x,y); hf==1 reads (z, next-x) and selects pad 0.
  v2f a;
  {
    const int baseq = q * 3 + 2 * hf;
    float a0 = cqb[baseq];
    float a1 = cqb[baseq + 1];
    a.x = a0;
    a.y = hf ? 0.0f : a1;
  }

  float bd[TKN];
  int   bi[TKN];
#pragma unroll
  for (int j = 0; j < TKN; ++j) { bd[j] = 3.4e38f; bi[j] = 0; }
  float worst = 3.4e38f;

  const int ntiles = N >> 4;
  for (int nt = 0; nt < ntiles; ++nt) {
    const int nk = nt * 16 + m;
    if (nt + 1 < ntiles) {
      __builtin_prefetch(&ckb[(nk + 16) * 3], 0, 3);  // global_prefetch_b8
    }
    v2f bb;
    {
      const int basek = nk * 3 + 2 * hf;
      float b0 = ckb[basek];
      float b1 = ckb[basek + 1];
      bb.x = b0;
      bb.y = hf ? 0.0f : b1;
    }

    v8f c = {0.f, 0.f, 0.f, 0.f, 0.f, 0.f, 0.f, 0.f};
    c = wmma_f32_16x16x4(a, bb, c);

    const float knv = knb[nk];
#pragma unroll
    for (int r = 0; r < 8; ++r) {
      tile[wave][r + 8 * hf][m] = knv - 2.0f * c[r];  // row M, col = key lane
    }
    lds_fence_wave();  // s_wait_dscnt: make tile visible across lanes of wave

    if (hf == 0) {
#pragma unroll
      for (int j = 0; j < 16; ++j) {
        float v = tile[wave][m][j];
        if (v < worst) {
          float mv = bd[0]; int mp = 0;
#pragma unroll
          for (int t = 1; t < TKN; ++t)
            if (bd[t] > mv) { mv = bd[t]; mp = t; }
#pragma unroll
          for (int t = 0; t < TKN; ++t)
            if (t == mp) { bd[t] = v; bi[t] = nt * 16 + j; }
          mv = bd[0];
#pragma unroll
          for (int t = 1; t < TKN; ++t) mv = fmaxf(mv, bd[t]);
          worst = mv;
        }
      }
    }
    lds_fence_wave();  // order reads before next tile's stores
  }

  if (hf == 0) {
    int* out = knn_idx + ((size_t)b * Q + q) * TKN;
#pragma unroll
    for (int j = 0; j < TKN; ++j) out[j] = bi[j];
  }
}

// ---------------------------------------------------------------------------
// Edge conv: y[b, q*16+k, :] = W @ concat(fk[nb]-fq, fq).
// One wave = 16 edge rows x 16 output channels; accumulate Cin2/4 WMMA steps.
// Channel loop split into the two concat halves -> branchless inner loops.
// ---------------------------------------------------------------------------
__global__ void k_edgeconv(const float* __restrict__ fq, const float* __restrict__ fk,
                           const int* __restrict__ knn_idx,
                           const float* __restrict__ W, float* __restrict__ y,
                           int Q, int N, int C, int Cout) {
  const int lane = threadIdx.x & 31;
  const int wave = threadIdx.x >> 5;
  const int gw = blockIdx.x * WAVES_PER_BLK + wave;
  const int ctiles = Cout >> 4;
  const int b = gw / (Q * ctiles);
  const int rem = gw % (Q * ctiles);
  const int rt = rem / ctiles;   // row tile == query index (K == 16)
  const int ct = rem % ctiles;
  const int m = lane & 15;
  const int hf = lane >> 4;

  const int qi = rt;
  const int kk = m;  // each lane's edge row = (qi, kk)
  const int nb = knn_idx[((size_t)b * Q + qi) * TKN + kk];
  const float* fqr = fq + ((size_t)b * Q + qi) * C;
  const float* fkr = fk + ((size_t)b * N + nb) * C;

  const int Cin2 = 2 * C;
  const int col = m;
  const int cout = ct * 16 + col;
  const float* Wr = W + (size_t)cout * Cin2;
  __builtin_prefetch(Wr, 0, 3);

  v8f acc = {0.f, 0.f, 0.f, 0.f, 0.f, 0.f, 0.f, 0.f};

  // First half: channels [0, C) -> e = fk[nb] - fq
  for (int c4 = 0; c4 < C; c4 += 4) {
    const int d0 = c4 + 2 * hf;
    v2f a;  a.x = fkr[d0] - fqr[d0]; a.y = fkr[d0 + 1] - fqr[d0 + 1];
    v2f bb; bb.x = Wr[d0];           bb.y = Wr[d0 + 1];
    acc = wmma_f32_16x16x4(a, bb, acc);
  }
  // Second half: channels [C, 2C) -> e = fq
  const float* Wr2 = Wr + C;
  for (int c4 = 0; c4 < C; c4 += 4) {
    const int d0 = c4 + 2 * hf;
    v2f a;  a.x = fqr[d0];  a.y = fqr[d0 + 1];
    v2f bb; bb.x = Wr2[d0]; bb.y = Wr2[d0 + 1];
    acc = wmma_f32_16x16x4(a, bb, acc);
  }

  float* yb = y + (((size_t)b * Q * TKN) + (size_t)rt * 16) * Cout + ct * 16;
#pragma unroll
  for (int r = 0; r < 8; ++r) {
    yb[(size_t)(r + 8 * hf) * Cout + col] = acc[r];
  }
}

// ---------------------------------------------------------------------------
// GroupNorm stats: one block per (b, g); deterministic grid-stride + LDS tree.
// stats[(b*G+g)*2] = mean, [..+1] = rsqrt(var + eps)
// ---------------------------------------------------------------------------
__global__ void k_gnstats(const float* __restrict__ y, float* __restrict__ stats,
                          int QK, int C, int G) {
  const int b = blockIdx.x / G;
  const int g = blockIdx.x % G;
  const int cg = C / G;
  const float* yb = y + (size_t)b * QK * C + (size_t)g * cg;
  const int total = QK * cg;
  float s = 0.f, s2 = 0.f;
  for (int i = threadIdx.x; i < total; i += blockDim.x) {
    const int r = i / cg, c = i % cg;
    const float v = yb[(size_t)r * C + c];
    s += v;
    s2 += v * v;
  }
  __shared__ float sh[256], sh2[256];
  sh[threadIdx.x] = s;
  sh2[threadIdx.x] = s2;
  __syncthreads();
  for (int off = 128; off > 0; off >>= 1) {
    if ((int)threadIdx.x < off) {
      sh[threadIdx.x] += sh[threadIdx.x + off];
      sh2[threadIdx.x] += sh2[threadIdx.x + off];
    }
    __syncthreads();
  }
  if (threadIdx.x == 0) {
    const float inv = 1.0f / (float)total;
    const float mean = sh[0] * inv;
    const float var = sh2[0] * inv - mean * mean;
    stats[blockIdx.x * 2 + 0] = mean;
    stats[blockIdx.x * 2 + 1] = rsqrtf(var + 1e-5f);
  }
}

// ---------------------------------------------------------------------------
// Normalize + affine + LeakyReLU(0.2) + max over K -> f (B, Q, C)
// ---------------------------------------------------------------------------
__global__ void k_gnmax(const float* __restrict__ y, const float* __restrict__ stats,
                        const float* __restrict__ gw, const float* __restrict__ gb,
                        float* __restrict__ f, int Q, int C, int G, int total) {
  const int i = blockIdx.x * blockDim.x + threadIdx.x;
  if (i >= total) return;
  const int c = i % C;
  const int q = (i / C) % Q;
  const int b = i / (C * Q);
  const int g = c / (C / G);
  const float mean = stats[(b * G + g) * 2 + 0];
  const float scale = stats[(b * G + g) * 2 + 1] * gw[c];
  const float bias = gb[c];
  const float* yr = y + ((size_t)b * Q * TKN + (size_t)q * TKN) * C + c;
  float best = -3.4e38f;
#pragma unroll
  for (int kk = 0; kk < TKN; ++kk) {
    float v = (yr[(size_t)kk * C] - mean) * scale + bias;
    v = (v > 0.f) ? v : 0.2f * v;
    best = fmaxf(best, v);
  }
  f[i] = best;
}

// ---------------------------------------------------------------------------
// Farthest point sampling: one block (256 thr) per batch. Register-resident
// min-dist, LDS argmax tree with first-occurrence (lowest index) tie-break.
// ---------------------------------------------------------------------------
template <int PPT>
__global__ void k_fps(const float* __restrict__ xyz, int* __restrict__ out,
                      int N, int n) {
  const int b = blockIdx.x;
  const float* xb = xyz + (size_t)b * N * 3;
  float dist[PPT];
#pragma unroll
  for (int i = 0; i < PPT; ++i) dist[i] = 3.4e38f;
  __shared__ float sv[256];
  __shared__ int si[256];
  __shared__ int sLast;
  int last = 0;
  if (threadIdx.x == 0) out[(size_t)b * n] = 0;
  for (int it = 1; it < n; ++it) {
    const float px = xb[last * 3 + 0];
    const float py = xb[last * 3 + 1];
    const float pz = xb[last * 3 + 2];
    float bv = -1.f;
    int bidx = 0x7fffffff;
#pragma unroll
    for (int i = 0; i < PPT; ++i) {
      const int p = (int)threadIdx.x + i * 256;
      const float dx = xb[p * 3 + 0] - px;
      const float dy = xb[p * 3 + 1] - py;
      const float dz = xb[p * 3 + 2] - pz;
      const float d = dx * dx + dy * dy + dz * dz;
      dist[i] = fminf(dist[i], d);
      if (dist[i] > bv || (dist[i] == bv && p < bidx)) { bv = dist[i]; bidx = p; }
    }
    sv[threadIdx.x] = bv;
    si[threadIdx.x] = bidx;
    __syncthreads();
    for (int off = 128; off > 0; off >>= 1) {
      if ((int)threadIdx.x < off) {
        const float v2 = sv[threadIdx.x + off];
        const int i2 = si[threadIdx.x + off];
        if (v2 > sv[threadIdx.x] ||
            (v2 == sv[threadIdx.x] && i2 < si[threadIdx.x])) {
          sv[threadIdx.x] = v2;
          si[threadIdx.x] = i2;
        }
      }
      __syncthreads();
    }
    if (threadIdx.x == 0) {
      sLast = si[0];
      out[(size_t)b * n + it] = si[0];
    }
    __syncthreads();
    last = sLast;
  }
}

// ---------------------------------------------------------------------------
// Gather coords / normals / plane / sqnorm / features by fps index.
// Feature rows copied as float4 (C is a multiple of 4; buffers 16B aligned).
// ---------------------------------------------------------------------------
__global__ void k_gather(const float* __restrict__ coorS,
                         const float* __restrict__ nrmS, int nrmStride,
                         const float* __restrict__ plS, int plStride,
                         const float* __restrict__ sqnS,
                         const float* __restrict__ fS, int C,
                         const int* __restrict__ idx, int N, int n,
                         float* __restrict__ coorD, float* __restrict__ nrmD,
                         float* __restrict__ plD, float* __restrict__ sqnD,
                         float* __restrict__ fD, int total) {
  const int i = blockIdx.x * blockDim.x + threadIdx.x;
  if (i >= total) return;
  const int b = i / n;
  const int s = idx[i];
  const size_t src = (size_t)b * N + s;
#pragma unroll
  for (int d = 0; d < 3; ++d) {
    coorD[(size_t)i * 3 + d] = coorS[src * 3 + d];
    nrmD[(size_t)i * 3 + d] = nrmS[src * nrmStride + d];
  }
  plD[i] = plS[src * plStride];
  sqnD[i] = sqnS[src];
  const float4* fs4 = (const float4*)(fS + src * C);
  float4* fd4 = (float4*)(fD + (size_t)i * C);
  for (int c = 0; c < (C >> 2); ++c) fd4[c] = fs4[c];
}

// ---------------------------------------------------------------------------
// Host orchestration
// ---------------------------------------------------------------------------
extern "C" void kernel_launch(void* const* d_in, const int* in_sizes, int n_in,
                              void* d_out, int out_size, void* d_ws, size_t ws_size,
                              hipStream_t stream) {
  (void)in_sizes; (void)n_in; (void)out_size; (void)ws_size;
  const int B = 8, N0 = 4096, Q1 = 1024, G = 4;

  const float* x   = (const float*)d_in[0];
  const float* Win = (const float*)d_in[1];
  const float* bin = (const float*)d_in[2];
  const float* W1  = (const float*)d_in[3];
  const float* g1w = (const float*)d_in[4];
  const float* g1b = (const float*)d_in[5];
  const float* W2  = (const float*)d_in[6];
  const float* g2w = (const float*)d_in[7];
  const float* g2b = (const float*)d_in[8];
  const float* W3  = (const float*)d_in[9];
  const float* g3w = (const float*)d_in[10];
  const float* g3b = (const float*)d_in[11];
  const float* W4  = (const float*)d_in[12];
  const float* g4w = (const float*)d_in[13];
  const float* g4b = (const float*)d_in[14];
  // d_in[15] = num0 (== 1024, fixed by setup)

  float* w = (float*)d_ws;
  size_t o = 0;
  auto alloc = [&](size_t nElem) { float* p = w + o; o += nElem; return p; };
  float* coor0 = alloc((size_t)B * N0 * 3);
  float* sqn0  = alloc((size_t)B * N0);
  float* f0    = alloc((size_t)B * N0 * 8);
  float* f1    = alloc((size_t)B * N0 * 32);
  float* coor1 = alloc((size_t)B * Q1 * 3);
  float* nrm1  = alloc((size_t)B * Q1 * 3);
  float* pl1   = alloc((size_t)B * Q1);
  float* sqn1  = alloc((size_t)B * Q1);
  float* f1q   = alloc((size_t)B * Q1 * 32);
  float* f2    = alloc((size_t)B * Q1 * 64);
  float* f3    = alloc((size_t)B * Q1 * 64);
  float* f3q   = alloc((size_t)B * Q1 * 64);
  float* sqn2  = alloc((size_t)B * Q1);
  float* stats = alloc(64);
  float* ybuf  = alloc((size_t)B * N0 * 16 * 32);  // largest y (reused per layer)
  int* knn_idx = (int*)alloc((size_t)B * N0 * 16);
  int* fps_idx = (int*)alloc((size_t)B * Q1);

  float* out      = (float*)d_out;
  float* out_coor = out;                       // (8,1024,3)
  float* out_f    = out + 24576;               // (8,1024,128)
  float* out_nrm  = out + 24576 + 1048576;     // (8,1024,3)
  float* out_pl   = out_nrm + 24576;           // (8,1024,1)

  const dim3 blk(256);

  // Stage 0: projection + norms
  k_proj<<<dim3((B * N0) / 256), blk, 0, stream>>>(x, Win, bin, coor0, f0, sqn0,
                                                   B * N0);

  // Stage 1: knn(4096x4096) + edgeconv W1 (Cin2=16, Cout=32) + GN + max
  k_knn<<<dim3(B * (N0 / 16) / WAVES_PER_BLK), blk, 0, stream>>>(
      coor0, coor0, sqn0, knn_idx, N0, N0);
  k_edgeconv<<<dim3(B * N0 * (32 / 16) / WAVES_PER_BLK), blk, 0, stream>>>(
      f0, f0, knn_idx, W1, ybuf, N0, N0, 8, 32);
  k_gnstats<<<dim3(B * G), blk, 0, stream>>>(ybuf, stats, N0 * 16, 32, G);
  k_gnmax<<<dim3((B * N0 * 32) / 256), blk, 0, stream>>>(
      ybuf, stats, g1w, g1b, f1, N0, 32, G, B * N0 * 32);

  // FPS 4096 -> 1024, gather (normals/plane straight from x, stride 7)
  k_fps<16><<<dim3(B), blk, 0, stream>>>(coor0, fps_idx, N0, Q1);
  k_gather<<<dim3((B * Q1) / 256), blk, 0, stream>>>(
      coor0, x + 3, 7, x + 6, 7, sqn0, f1, 32, fps_idx, N0, Q1, coor1, nrm1,
      pl1, sqn1, f1q, B * Q1);

  // Stage 2: knn(1024 queries vs 4096 keys) + edgeconv W2 (Cin2=64, Cout=64)
  k_knn<<<dim3(B * (Q1 / 16) / WAVES_PER_BLK), blk, 0, stream>>>(
      coor1, coor0, sqn0, knn_idx, Q1, N0);
  k_edgeconv<<<dim3(B * Q1 * (64 / 16) / WAVES_PER_BLK), blk, 0, stream>>>(
      f1q, f1, knn_idx, W2, ybuf, Q1, N0, 32, 64);
  k_gnstats<<<dim3(B * G), blk, 0, stream>>>(ybuf, stats, Q1 * 16, 64, G);
  k_gnmax<<<dim3((B * Q1 * 64) / 256), blk, 0, stream>>>(
      ybuf, stats, g2w, g2b, f2, Q1, 64, G, B * Q1 * 64);

  // Stage 3: knn(1024x1024) + edgeconv W3 (Cin2=128, Cout=64)
  k_knn<<<dim3(B * (Q1 / 16) / WAVES_PER_BLK), blk, 0, stream>>>(
      coor1, coor1, sqn1, knn_idx, Q1, Q1);
  k_edgeconv<<<dim3(B * Q1 * (64 / 16) / WAVES_PER_BLK), blk, 0, stream>>>(
      f2, f2, knn_idx, W3, ybuf, Q1, Q1, 64, 64);
  k_gnstats<<<dim3(B * G), blk, 0, stream>>>(ybuf, stats, Q1 * 16, 64, G);
  k_gnmax<<<dim3((B * Q1 * 64) / 256), blk, 0, stream>>>(
      ybuf, stats, g3w, g3b, f3, Q1, 64, G, B * Q1 * 64);

  // FPS 1024 -> 1024 (reorders), gather into final output slices
  k_fps<4><<<dim3(B), blk, 0, stream>>>(coor1, fps_idx, Q1, Q1);
  k_gather<<<dim3((B * Q1) / 256), blk, 0, stream>>>(
      coor1, nrm1, 3, pl1, 1, sqn1, f3, 64, fps_idx, Q1, Q1, out_coor, out_nrm,
      out_pl, sqn2, f3q, B * Q1);

  // Stage 4: knn(reordered 1024 vs 1024) + edgeconv W4 (Cin2=128, Cout=128)
  k_knn<<<dim3(B * (Q1 / 16) / WAVES_PER_BLK), blk, 0, stream>>>(
      out_coor, coor1, sqn1, knn_idx, Q1, Q1);
  k_edgeconv<<<dim3(B * Q1 * (128 / 16) / WAVES_PER_BLK), blk, 0, stream>>>(
      f3q, f3, knn_idx, W4, ybuf, Q1, Q1, 64, 128);
  k_gnstats<<<dim3(B * G), blk, 0, stream>>>(ybuf, stats, Q1 * 16, 128, G);
  k_gnmax<<<dim3((B * Q1 * 128) / 256), blk, 0, stream>>>(
      ybuf, stats, g4w, g4b, out_f, Q1, 128, G, B * Q1 * 128);
}